// VectorizedQuantumCircuit_74474732913026
// MI455X (gfx1250) — compile-verified
//
#include <hip/hip_runtime.h>

#define NQ      14
#define DIM     16384          // 2^14
#define BATCH   256
#define THREADS 256            // 8 wave32 per workgroup

typedef float v2f __attribute__((ext_vector_type(2)));
typedef float v8f __attribute__((ext_vector_type(8)));

// Build U = RY(a0) (x) RY(a1) (x) RY(a2) (x) RY(a3)  (16x16, real).
// RY(a) = [[cos(a/2), -sin(a/2)], [sin(a/2), cos(a/2)]]
// One thread per entry (256 threads, 256 entries).
__device__ __forceinline__ void compute_ry_u(float* __restrict__ U,
                                             float a0, float a1, float a2, float a3,
                                             int tid) {
    const int i = tid >> 4;      // row
    const int j = tid & 15;      // col
    float ang[4] = {a0, a1, a2, a3};
    float v = 1.0f;
#pragma unroll
    for (int t = 0; t < 4; ++t) {
        float s, c;
        __sincosf(ang[t] * 0.5f, &s, &c);
        const int ib = (i >> (3 - t)) & 1;
        const int jb = (j >> (3 - t)) & 1;
        v *= (ib == jb) ? c : (ib ? s : -s);   // M[ib][jb]
    }
    U[tid] = v;
}

// Apply real 16x16 matrix U to the 4-qubit group whose lowest bit position is s.
// State viewed as (h, j, low): index = (h << (s+4)) + (j << s) + low, j in [0,16).
// new[h,i,low] = sum_j U[i][j] * state[h,j,low]  -> D = U x S via WMMA f32 16x16x4.
__device__ __forceinline__ void sweep_group(float* __restrict__ re,
                                            float* __restrict__ im,
                                            const float* __restrict__ U,
                                            int s, int tid) {
    const int lane = tid & 31;
    const int wave = tid >> 5;
    const int m    = lane & 15;
    const int koff = (lane >> 4) << 1;      // lanes 0-15: K 0,1 ; lanes 16-31: K 2,3

    // A fragments: U rows, K chunks of 4 (loaded once, reused for all tiles)
    v2f A[4];
#pragma unroll
    for (int kk = 0; kk < 4; ++kk) {
        A[kk][0] = U[m * 16 + kk * 4 + koff + 0];
        A[kk][1] = U[m * 16 + kk * 4 + koff + 1];
    }

    const int jstr    = 1 << s;
    const int lowmask = jstr - 1;

    // 1024 columns total -> 64 tiles of 16 columns; 8 waves -> 8 tiles per wave.
    for (int t = 0; t < 8; ++t) {
        const int col  = (t * 8 + wave) * 16 + m;                 // this lane's column
        const int base = ((col >> s) << (s + 4)) + (col & lowmask);

        v8f aR = {};
        v8f aI = {};
#pragma unroll
        for (int kk = 0; kk < 4; ++kk) {
            const int j0 = kk * 4 + koff;
            v2f Br, Bi;
            Br[0] = re[base + (j0 + 0) * jstr];
            Br[1] = re[base + (j0 + 1) * jstr];
            Bi[0] = im[base + (j0 + 0) * jstr];
            Bi[1] = im[base + (j0 + 1) * jstr];
            // (neg_a, A, neg_b, B, c_mod, C, reuse_a, reuse_b)
            aR = __builtin_amdgcn_wmma_f32_16x16x4_f32(false, A[kk], false, Br,
                                                       (short)0, aR, false, false);
            aI = __builtin_amdgcn_wmma_f32_16x16x4_f32(false, A[kk], false, Bi,
                                                       (short)0, aI, false, false);
        }

        // D layout: VGPR v holds row v (lanes 0-15) / row v+8 (lanes 16-31), col = lane&15
        const int r0 = (lane >> 4) << 3;
#pragma unroll
        for (int v = 0; v < 8; ++v) {
            re[base + (r0 + v) * jstr] = aR[v];
            im[base + (r0 + v) * jstr] = aI[v];
        }
    }
}

__global__ void __launch_bounds__(THREADS)
qcsim_kernel(const float* __restrict__ x,       // (256,14) encoding angles
             const float* __restrict__ t0y,     // (3,14) RY layer angles
             const float* __restrict__ t1z,     // (3,14) RZ layer angles
             float2* __restrict__ out) {        // (256,16384) complex64
    extern __shared__ float smem[];
    float* re = smem;              // 16384 floats
    float* im = smem + DIM;        // 16384 floats
    float* U  = smem + 2 * DIM;    // 256 floats

    const int b   = blockIdx.x;
    const int tid = threadIdx.x;
    const float* xb = x + b * NQ;

    // |0...0>
    for (int k = tid; k < DIM; k += THREADS) {
        re[k] = (k == 0) ? 1.0f : 0.0f;
        im[k] = 0.0f;
    }
    __syncthreads();

    // Qubit q lives at bit (13 - q). Group low-bit positions:
    // {q0-3}: s=10, {q4-7}: s=6, {q8-11}: s=2, {q10-13 w/ identity on 10,11}: s=0
    const int sgrp[4] = {10, 6, 2, 0};

    // ---------------- Encoding: RY(x_q) on every qubit (4 fused sweeps) -------
    for (int g = 0; g < 4; ++g) {
        float a0, a1, a2, a3;
        if (g < 3) { a0 = xb[g*4]; a1 = xb[g*4+1]; a2 = xb[g*4+2]; a3 = xb[g*4+3]; }
        else       { a0 = 0.0f;   a1 = 0.0f;      a2 = xb[12];    a3 = xb[13]; }
        compute_ry_u(U, a0, a1, a2, a3, tid);
        __syncthreads();
        sweep_group(re, im, U, sgrp[g], tid);
        __syncthreads();
    }

    // ---------------- Layers --------------------------------------------------
    for (int layer = 0; layer < 3; ++layer) {
        // RY(theta_0_y[layer, q]) on every qubit: 4 fused WMMA sweeps
        const float* th = t0y + layer * NQ;
        for (int g = 0; g < 4; ++g) {
            float a0, a1, a2, a3;
            if (g < 3) { a0 = th[g*4]; a1 = th[g*4+1]; a2 = th[g*4+2]; a3 = th[g*4+3]; }
            else       { a0 = 0.0f;    a1 = 0.0f;      a2 = th[12];    a3 = th[13]; }
            compute_ry_u(U, a0, a1, a2, a3, tid);
            __syncthreads();
            sweep_group(re, im, U, sgrp[g], tid);
            __syncthreads();
        }

        // Circular CNOT ring: CNOT(q, q+1 mod 14), in-place LDS pair swaps
        for (int q = 0; q < NQ; ++q) {
            const int c  = q;
            const int tq = (q + 1) % NQ;
            const int pc = 13 - c;              // control bit position
            const int pt = 13 - tq;             // target bit position
            const int plo = pc < pt ? pc : pt;
            const int phi = pc < pt ? pt : pc;
            for (int p = tid; p < DIM / 4; p += THREADS) {
                // insert 0-bits at plo then phi into the 12-bit pair id
                uint32_t v  = (uint32_t)p;
                uint32_t t1 = ((v >> plo) << (plo + 1)) | (v & ((1u << plo) - 1));
                uint32_t t2 = ((t1 >> phi) << (phi + 1)) | (t1 & ((1u << phi) - 1));
                uint32_t i0 = t2 | (1u << pc);          // control=1, target=0
                uint32_t i1 = i0 | (1u << pt);          // control=1, target=1
                float tr = re[i0]; re[i0] = re[i1]; re[i1] = tr;
                float ti = im[i0]; im[i0] = im[i1]; im[i1] = ti;
            }
            __syncthreads();
        }

        // RZ(theta_1_z[layer, q]) on every qubit: diagonal -> elementwise phase
        const float* tzp = t1z + layer * NQ;
        float tz[NQ];
        float hs = 0.0f;
#pragma unroll
        for (int q = 0; q < NQ; ++q) { tz[q] = tzp[q]; hs += tz[q]; }
        hs *= 0.5f;
        for (int k = tid; k < DIM; k += THREADS) {
            // phase = sum_{bit set} theta_q - 0.5 * sum theta_q
            float phi_ang = -hs;
#pragma unroll
            for (int q = 0; q < NQ; ++q)
                if ((k >> (13 - q)) & 1) phi_ang += tz[q];
            float sn, cs;
            __sincosf(phi_ang, &sn, &cs);
            const float r = re[k], i2 = im[k];
            re[k] = r * cs - i2 * sn;
            im[k] = r * sn + i2 * cs;
        }
        __syncthreads();
    }

    // ---------------- Store final state (interleaved complex64) ---------------
    float2* ob = out + (size_t)b * DIM;
    for (int k = tid; k < DIM; k += THREADS) {
        ob[k] = make_float2(re[k], im[k]);
    }
}

extern "C" void kernel_launch(void* const* d_in, const int* in_sizes, int n_in,
                              void* d_out, int out_size, void* d_ws, size_t ws_size,
                              hipStream_t stream) {
    (void)in_sizes; (void)n_in; (void)out_size; (void)d_ws; (void)ws_size;
    const float* x   = (const float*)d_in[0];   // (256,14)
    const float* t0y = (const float*)d_in[1];   // (3,14)
    const float* t1z = (const float*)d_in[2];   // (3,14)
    float2* out = (float2*)d_out;               // (256,16384) complex64

    const size_t smem = (size_t)(2 * DIM + 256) * sizeof(float); // 132 KB < 320 KB WGP LDS
    hipFuncSetAttribute(reinterpret_cast<const void*>(qcsim_kernel),
                        hipFuncAttributeMaxDynamicSharedMemorySize, (int)smem);
    qcsim_kernel<<<BATCH, THREADS, smem, stream>>>(x, t0y, t1z, out);
}